// GraphConv_80444737454173
// MI455X (gfx1250) — compile-verified
//
#include <hip/hip_runtime.h>
#include <hip/hip_bf16.h>

// GraphConv fused kernel for MI455X (gfx1250, wave32).
// out[b,i,u] = (sum_j edges[b,i,j] * h[b,j,u]) / (sum_j edges[b,i,j]),
// h = X W + bias.  Memory-bound on the single 256MB streaming pass over
// `edges`; fp32 WMMA (V_WMMA_F32_16X16X4_F32) supplies the ~373 TFLOP/s
// needed to keep pace with 23.3 TB/s HBM.  den is fused into the same pass.

typedef __attribute__((ext_vector_type(2))) float v2f;
typedef __attribute__((ext_vector_type(8))) float v8f;

namespace {
constexpr int kB = 4;
constexpr int kN = 4096;
constexpr int kF = 32;
constexpr int kU = 32;
}

// ---------------------------------------------------------------------------
// Stage 1: hT[b, u, n] = sum_f X[b,n,f] * W[f,u] + bias[u]   (transposed h)
// Transposed so stage-2 B-fragments are contiguous b64 loads along n (=K).
// Block: 256 threads = 32 u-lanes x 8 n-rows; W and an X chunk staged in LDS.
// ---------------------------------------------------------------------------
__global__ __launch_bounds__(256) void hT_kernel(const float* __restrict__ X,
                                                 const float* __restrict__ W,
                                                 const float* __restrict__ bias,
                                                 float* __restrict__ hT) {
  __shared__ float Wlds[kF][kU];   // 4 KB
  __shared__ float Xlds[8][kF];    // 1 KB

  const int b  = blockIdx.x / (kN / 8);
  const int n0 = (blockIdx.x % (kN / 8)) * 8;

  // Cooperative loads
  for (int idx = threadIdx.x; idx < kF * kU; idx += 256) {
    Wlds[idx / kU][idx % kU] = W[idx];
  }
  {
    const int row = threadIdx.x / kF;      // 0..7
    const int f   = threadIdx.x % kF;      // 0..31
    Xlds[row][f] = X[((size_t)(b * kN + n0 + row)) * kF + f];
  }
  __syncthreads();

  const int u  = threadIdx.x & 31;
  const int nn = threadIdx.x >> 5;         // 0..7
  float acc = bias[u];
#pragma unroll
  for (int f = 0; f < kF; ++f) {
    acc += Xlds[nn][f] * Wlds[f][u];
  }
  hT[((size_t)(b * kU + u)) * kN + (n0 + nn)] = acc;
}

// ---------------------------------------------------------------------------
// Stage 2: one workgroup per (batch, 16-row block). 4 waves split K=4096 into
// 1024-wide chunks.  fp32 WMMA fragment layouts (ISA 7.12.2, half-offset):
//   A 16x4 : lane l<16 -> M=l, VGPR{0,1}=K{0,1}; lanes 16+l -> K{2,3}
//   B 4x16 : lane l<16 -> N=l, VGPR{0,1}=K{0,1}; lanes 16+l -> K{2,3}
//   C 16x16: VGPR r, lanes 0-15 -> M=r, lanes 16-31 -> M=r+8
// So both A and B fragments are contiguous float2 loads at column j + 2*hi.
// ---------------------------------------------------------------------------
__global__ __launch_bounds__(128) void graphconv_wmma_kernel(
    const float* __restrict__ edges,   // [B, N, N]
    const float* __restrict__ hT,      // [B, U, N]
    float* __restrict__ out) {         // [B, N, U]
  __shared__ float red[4][16][kU];     // per-wave partial 16x32 tiles (8 KB)
  __shared__ float denred[4][32];      // per-wave per-lane row-sum partials

  const int b    = blockIdx.x >> 8;          // / 256
  const int i0   = (blockIdx.x & 255) << 4;  // * 16
  const int wave = threadIdx.x >> 5;
  const int lane = threadIdx.x & 31;
  const int l    = lane & 15;
  const int hi   = lane >> 4;
  const int off  = 2 * hi;

  const float* __restrict__ arow = edges + ((size_t)(b * kN + i0 + l)) * kN;
  const float* __restrict__ h0   = hT + ((size_t)(b * kU + l)) * kN;
  const float* __restrict__ h1   = hT + ((size_t)(b * kU + 16 + l)) * kN;

  v8f acc0 = {};   // u = 0..15
  v8f acc1 = {};   // u = 16..31
  float den = 0.0f;

  const int jbeg = wave * (kN / 4);
  const int jend = jbeg + (kN / 4);
#pragma unroll 4
  for (int j = jbeg; j < jend; j += 4) {
    // edges streamed once: non-temporal so the 256MB stream doesn't evict hT
    v2f a  = __builtin_nontemporal_load((const v2f*)(arow + j + off));
    v2f b0 = *(const v2f*)(h0 + j + off);
    v2f b1 = *(const v2f*)(h1 + j + off);
    den += a.x + a.y;   // fused row-sum (avoids a second 256MB pass)
    acc0 = __builtin_amdgcn_wmma_f32_16x16x4_f32(
        false, a, false, b0, (short)0, acc0, false, false);
    acc1 = __builtin_amdgcn_wmma_f32_16x16x4_f32(
        false, a, false, b1, (short)0, acc1, false, false);
  }

  // Stash per-wave partials to LDS (C layout: VGPR r -> row r / r+8).
#pragma unroll
  for (int r = 0; r < 8; ++r) {
    red[wave][r + hi * 8][l]      = acc0[r];
    red[wave][r + hi * 8][16 + l] = acc1[r];
  }
  denred[wave][lane] = den;
  __syncthreads();

  // Cross-wave reduce, divide by den, coalesced store. 512 elems / 128 thr.
#pragma unroll
  for (int k = 0; k < 4; ++k) {
    const int e = threadIdx.x + 128 * k;
    const int m = e >> 5;        // output row within block (0..15)
    const int u = e & 31;        // output feature
    float num = red[0][m][u] + red[1][m][u] + red[2][m][u] + red[3][m][u];
    float d = denred[0][m] + denred[0][m + 16] +
              denred[1][m] + denred[1][m + 16] +
              denred[2][m] + denred[2][m + 16] +
              denred[3][m] + denred[3][m + 16];
    out[((size_t)(b * kN + i0 + m)) * kU + u] = num / d;
  }
}

extern "C" void kernel_launch(void* const* d_in, const int* in_sizes, int n_in,
                              void* d_out, int out_size, void* d_ws, size_t ws_size,
                              hipStream_t stream) {
  (void)in_sizes; (void)n_in; (void)out_size; (void)ws_size;
  const float* features = (const float*)d_in[0];  // [B, N, F]
  const float* edges    = (const float*)d_in[1];  // [B, N, N]
  const float* W        = (const float*)d_in[2];  // [F, U]
  const float* bias     = (const float*)d_in[3];  // [U]
  float* out = (float*)d_out;                     // [B, N, U]
  float* hT  = (float*)d_ws;                      // [B, U, N] = 2 MB scratch

  hT_kernel<<<kB * (kN / 8), 256, 0, stream>>>(features, W, bias, hT);
  graphconv_wmma_kernel<<<kB * (kN / 16), 128, 0, stream>>>(edges, hT, out);
}